// ContrastiveNoAugLoss_6184752906727
// MI455X (gfx1250) — compile-verified
//
#include <hip/hip_runtime.h>
#include <math.h>

typedef __attribute__((ext_vector_type(2))) float v2f;
typedef __attribute__((ext_vector_type(8))) float v8f;

#define B 128
#define N_ELEM 3072     // 3*32*32
#define N_PAD 4096
#define EPS 1e-12f

// ---------------- Kernel 1: per-row bitonic sort of x into xs ----------------
// One block per row, 1024 threads, 16KB LDS. Pad to 4096 with +INF (ascending).
__global__ void __launch_bounds__(1024)
sort_rows_kernel(const float* __restrict__ x, float* __restrict__ xs) {
    __shared__ float s[N_PAD];
    const int row = blockIdx.x;
    const int tid = threadIdx.x;

    for (int i = tid; i < N_PAD; i += 1024)
        s[i] = (i < N_ELEM) ? x[row * N_ELEM + i] : 3.402823e38f;
    __syncthreads();

    for (int k = 2; k <= N_PAD; k <<= 1) {
        for (int j = k >> 1; j > 0; j >>= 1) {
            for (int i = tid; i < N_PAD; i += 1024) {
                int ixj = i ^ j;
                if (ixj > i) {
                    float a = s[i], b2 = s[ixj];
                    bool up = ((i & k) == 0);
                    if ((a > b2) == up) { s[i] = b2; s[ixj] = a; }
                }
            }
            __syncthreads();
        }
    }

    for (int i = tid; i < N_ELEM; i += 1024)
        xs[row * N_ELEM + i] = s[i];
}

// ---------------- Kernel 2: L2-normalize rows of z -> zn ----------------
// One block per row, 128 threads (4 waves), shuffle+LDS reduction.
__global__ void __launch_bounds__(128)
znorm_kernel(const float* __restrict__ z, float* __restrict__ zn) {
    const int row = blockIdx.x;
    const int tid = threadIdx.x;
    float v = z[row * B + tid];
    float s = v * v;
    #pragma unroll
    for (int off = 16; off > 0; off >>= 1) s += __shfl_xor(s, off, 32);
    __shared__ float wsum[4];
    if ((tid & 31) == 0) wsum[tid >> 5] = s;
    __syncthreads();
    float total = wsum[0] + wsum[1] + wsum[2] + wsum[3];
    float nrm = fmaxf(sqrtf(total), EPS);
    zn[row * B + tid] = v / nrm;
}

// ---------------- Kernel 3: z_cos = zn @ zn^T via V_WMMA_F32_16X16X4_F32 ----
// 64 output tiles (8x8 grid of 16x16 tiles). One tile per wave.
// 16 blocks x 128 threads = 64 waves. K=128 -> 32 wmma ops per wave.
// A-frag layout (ISA 7.12.2, 32-bit A 16x4): lane 0-15 hold M=lane with
// VGPR0=K0,VGPR1=K1; lanes 16-31 hold M=lane-16 with VGPR0=K2,VGPR1=K3.
// B = zn^T, so the B fragment load mirrors A with the N-tile index.
__global__ void __launch_bounds__(128)
zcos_wmma_kernel(const float* __restrict__ zn, float* __restrict__ zcos) {
    const int lane   = threadIdx.x & 31;
    const int waveIb = threadIdx.x >> 5;
    const int waveG  = blockIdx.x * 4 + waveIb;   // 0..63
    const int tm = waveG >> 3;                    // 0..7
    const int tn = waveG & 7;                     // 0..7
    const int half = lane >> 4;                   // 0 or 1
    const int r    = lane & 15;

    const float* __restrict__ Arow = zn + (tm * 16 + r) * B;
    const float* __restrict__ Brow = zn + (tn * 16 + r) * B;

    v8f c = {};
    #pragma unroll 4
    for (int k0 = 0; k0 < B; k0 += 4) {
        v2f a, b;
        a.x = Arow[k0 + 2 * half + 0];
        a.y = Arow[k0 + 2 * half + 1];
        b.x = Brow[k0 + 2 * half + 0];
        b.y = Brow[k0 + 2 * half + 1];
        c = __builtin_amdgcn_wmma_f32_16x16x4_f32(
                /*neg_a=*/false, a, /*neg_b=*/false, b,
                /*c_mod=*/(short)0, c, /*reuse_a=*/false, /*reuse_b=*/false);
    }

    // C/D layout: VGPR v: lanes 0-15 -> M=v, N=lane; lanes 16-31 -> M=v+8.
    #pragma unroll
    for (int v = 0; v < 8; ++v) {
        int m = tm * 16 + v + 8 * half;
        int n = tn * 16 + r;
        zcos[m * B + n] = c[v];
    }
}

// ---------------- Kernel 4: zmax = max(z_cos) ----------------
__global__ void __launch_bounds__(256)
zmax_kernel(const float* __restrict__ zcos, float* __restrict__ zmax) {
    const int tid = threadIdx.x;
    float m = -3.402823e38f;
    for (int i = tid; i < B * B; i += 256) m = fmaxf(m, zcos[i]);
    #pragma unroll
    for (int off = 16; off > 0; off >>= 1) m = fmaxf(m, __shfl_xor(m, off, 32));
    __shared__ float wm[8];
    if ((tid & 31) == 0) wm[tid >> 5] = m;
    __syncthreads();
    if (tid == 0) {
        float mm = wm[0];
        #pragma unroll
        for (int w = 1; w < 8; ++w) mm = fmaxf(mm, wm[w]);
        *zmax = mm;
    }
}

// ---------------- Kernel 5: pairwise 1D-EMD on sorted rows ----------------
// Block per row i (row i cached in LDS); 8 waves per block each own
// j = wave, wave+8, ... ; per-wave shuffle reduction; float4 streamed loads.
__global__ void __launch_bounds__(256)
emd_kernel(const float* __restrict__ xs, float* __restrict__ xemd) {
    __shared__ __align__(16) float xi[N_ELEM];
    const int i   = blockIdx.x;
    const int tid = threadIdx.x;
    for (int k = tid; k < N_ELEM; k += 256) xi[k] = xs[i * N_ELEM + k];
    __syncthreads();

    const int w = tid >> 5;
    const int l = tid & 31;
    const float4* __restrict__ xi4 = reinterpret_cast<const float4*>(xi);

    for (int j = w; j < B; j += 8) {
        const float4* __restrict__ xj4 =
            reinterpret_cast<const float4*>(xs + j * N_ELEM);
        float s = 0.f;
        for (int k4 = l; k4 < N_ELEM / 4; k4 += 32) {
            float4 a = xi4[k4];
            float4 b = xj4[k4];
            s += fabsf(a.x - b.x) + fabsf(a.y - b.y) +
                 fabsf(a.z - b.z) + fabsf(a.w - b.w);
        }
        #pragma unroll
        for (int off = 16; off > 0; off >>= 1) s += __shfl_xor(s, off, 32);
        if (l == 0) xemd[i * B + j] = s * (1.0f / (float)N_ELEM);
    }
}

// ---------------- Kernel 6: loss = mean((x_emd - (zmax - z_cos))^2) ---------
__global__ void __launch_bounds__(256)
loss_kernel(const float* __restrict__ xemd, const float* __restrict__ zcos,
            const float* __restrict__ zmaxp, float* __restrict__ out) {
    const int tid = threadIdx.x;
    const float zmax = *zmaxp;
    float acc = 0.f;
    for (int i = tid; i < B * B; i += 256) {
        float d = xemd[i] - (zmax - zcos[i]);
        acc += d * d;
    }
    #pragma unroll
    for (int off = 16; off > 0; off >>= 1) acc += __shfl_xor(acc, off, 32);
    __shared__ float wsum[8];
    if ((tid & 31) == 0) wsum[tid >> 5] = acc;
    __syncthreads();
    if (tid == 0) {
        float t = 0.f;
        #pragma unroll
        for (int w = 0; w < 8; ++w) t += wsum[w];
        out[0] = t * (1.0f / (float)(B * B));
    }
}

// ---------------- Host-side launch ----------------
extern "C" void kernel_launch(void* const* d_in, const int* in_sizes, int n_in,
                              void* d_out, int out_size, void* d_ws, size_t ws_size,
                              hipStream_t stream) {
    const float* z = (const float*)d_in[0];   // [128,128]
    const float* x = (const float*)d_in[1];   // [128,3,32,32]
    float* out = (float*)d_out;               // scalar

    char* ws = (char*)d_ws;
    float* xs   = (float*)(ws);                               // 128*3072 f32
    float* zn   = (float*)(ws + (size_t)B * N_ELEM * 4);      // 128*128 f32
    float* zcos = zn   + B * B;
    float* xemd = zcos + B * B;
    float* zmax = xemd + B * B;

    sort_rows_kernel<<<B, 1024, 0, stream>>>(x, xs);
    znorm_kernel<<<B, 128, 0, stream>>>(z, zn);
    zcos_wmma_kernel<<<16, 128, 0, stream>>>(zn, zcos);
    zmax_kernel<<<1, 256, 0, stream>>>(zcos, zmax);
    emd_kernel<<<B, 256, 0, stream>>>(xs, xemd);
    loss_kernel<<<1, 256, 0, stream>>>(xemd, zcos, zmax, out);
}